// GcnLayers_9569187136139
// MI455X (gfx1250) — compile-verified
//
#include <hip/hip_runtime.h>
#include <hip/hip_bf16.h>

// GCN: x -> [GEMM fp32 WMMA] -> fts -> [edge gather/scale/scatter-add] -> +b, PReLU
// N=100000, E=3200000, D=256, 3 layers.

typedef __attribute__((ext_vector_type(2))) float v2f;
typedef __attribute__((ext_vector_type(8))) float v8f;

#define GCN_D 256
#define LDS_STRIDE 260     // 260 % 64banks = 4 -> lanes 0..15 hit distinct banks; keeps 8B/16B align
#define ROWS_PER_BLOCK 64  // 4 row tiles per wave: amortizes W (B-fragment) loads 4x

// ---------------------------------------------------------------------------
// GEMM: F[n x 256] = X[n x 256] @ W[256 x 256], fp32, v_wmma_f32_16x16x4_f32.
// Block = 256 threads (8 waves). Block stages a 64-row strip of X in LDS.
// Wave w owns column tiles 2w, 2w+1 and all 4 row tiles (8 accumulators).
// Per k-step: 2 B-fragments (4 b32 loads, W hot in L2) reused across 4 A-frags
// (conflict-free ds b64) -> 8 WMMAs. 0.5 global loads per WMMA.
// ---------------------------------------------------------------------------
__global__ __launch_bounds__(256) void gcn_gemm_wmma(const float* __restrict__ X,
                                                     const float* __restrict__ W,
                                                     float* __restrict__ F,
                                                     int n_rows) {
    __shared__ float As[ROWS_PER_BLOCK * LDS_STRIDE];

    const int tid = threadIdx.x;
    const int row_base = blockIdx.x * ROWS_PER_BLOCK;

    // Cooperative, fully-coalesced load of the 64x256 X strip into LDS (float4).
    for (int v = tid; v < ROWS_PER_BLOCK * (GCN_D / 4); v += 256) {
        const int row = v >> 6;        // 64 float4 per row
        const int c4  = v & 63;
        int grow = row_base + row;
        if (grow >= n_rows) grow = n_rows - 1;  // clamp (tail safety)
        const float4 d = *reinterpret_cast<const float4*>(X + (size_t)grow * GCN_D + c4 * 4);
        *reinterpret_cast<float4*>(&As[row * LDS_STRIDE + c4 * 4]) = d;
    }
    __syncthreads();

    const int wave = tid >> 5;
    const int lane = tid & 31;
    const int j    = lane & 15;   // N (and M) index within tile
    const int half = lane >> 4;   // lane group: selects K pair / M+8

    const int col0 = wave * 32;
    const int col1 = wave * 32 + 16;

    v8f acc[4][2] = {};

    for (int k = 0; k < GCN_D; k += 4) {
        const int kk = k + 2 * half;
        // B fragments (4x16): lane(j,half) holds W[kk][col+j], W[kk+1][col+j].
        v2f b0, b1;
        b0[0] = W[(size_t)kk * GCN_D + col0 + j];
        b0[1] = W[(size_t)(kk + 1) * GCN_D + col0 + j];
        b1[0] = W[(size_t)kk * GCN_D + col1 + j];
        b1[1] = W[(size_t)(kk + 1) * GCN_D + col1 + j];
#pragma unroll
        for (int rt = 0; rt < 4; ++rt) {
            // A fragment (16x4) for row tile rt, bank-conflict-free b64 from LDS.
            v2f a = *reinterpret_cast<const v2f*>(&As[(rt * 16 + j) * LDS_STRIDE + kk]);
            acc[rt][0] = __builtin_amdgcn_wmma_f32_16x16x4_f32(false, a, false, b0, (short)0,
                                                               acc[rt][0], false, false);
            acc[rt][1] = __builtin_amdgcn_wmma_f32_16x16x4_f32(false, a, false, b1, (short)0,
                                                               acc[rt][1], false, false);
        }
    }

    // D layout: VGPR r, lanes 0-15 -> M=r, lanes 16-31 -> M=8+r.
    if (row_base + ROWS_PER_BLOCK <= n_rows) {
        // Full block: unguarded coalesced stores (common case).
#pragma unroll
        for (int rt = 0; rt < 4; ++rt) {
#pragma unroll
            for (int r = 0; r < 8; ++r) {
                const int row = row_base + rt * 16 + r + 8 * half;
                F[(size_t)row * GCN_D + col0 + j] = acc[rt][0][r];
                F[(size_t)row * GCN_D + col1 + j] = acc[rt][1][r];
            }
        }
    } else {
        // Tail block only.
#pragma unroll
        for (int rt = 0; rt < 4; ++rt) {
#pragma unroll
            for (int r = 0; r < 8; ++r) {
                const int row = row_base + rt * 16 + r + 8 * half;
                if (row < n_rows) {
                    F[(size_t)row * GCN_D + col0 + j] = acc[rt][0][r];
                    F[(size_t)row * GCN_D + col1 + j] = acc[rt][1][r];
                }
            }
        }
    }
}

// ---------------------------------------------------------------------------
// Zero the accumulation buffer (float4 stores).
// ---------------------------------------------------------------------------
__global__ __launch_bounds__(256) void gcn_zero(float* __restrict__ C, int total4) {
    const int i = blockIdx.x * blockDim.x + threadIdx.x;
    if (i < total4) {
        float4 z = {0.f, 0.f, 0.f, 0.f};
        *reinterpret_cast<float4*>(C + (size_t)i * 4) = z;
    }
}

// ---------------------------------------------------------------------------
// SpMM scatter: one wave32 per edge. Lane l owns features [8l, 8l+8):
// two coalesced b128 gathers of fts[col] (L2-resident: 102 MB < 192 MB L2),
// then 8 non-returning global_atomic_add_f32 into C[row].
// ---------------------------------------------------------------------------
__global__ __launch_bounds__(256) void gcn_spmm(const int* __restrict__ erows,
                                                const int* __restrict__ ecols,
                                                const float* __restrict__ evals,
                                                const float* __restrict__ F,
                                                float* __restrict__ C,
                                                int n_edges) {
    const int e    = blockIdx.x * 8 + (threadIdx.x >> 5);
    const int lane = threadIdx.x & 31;
    if (e >= n_edges) return;

    const int   r = erows[e];
    const int   c = ecols[e];
    const float v = evals[e];

    const float* src = F + (size_t)c * GCN_D + lane * 8;
    const float4 f0 = *reinterpret_cast<const float4*>(src);
    const float4 f1 = *reinterpret_cast<const float4*>(src + 4);

    float* dst = C + (size_t)r * GCN_D + lane * 8;
    atomicAdd(dst + 0, v * f0.x);
    atomicAdd(dst + 1, v * f0.y);
    atomicAdd(dst + 2, v * f0.z);
    atomicAdd(dst + 3, v * f0.w);
    atomicAdd(dst + 4, v * f1.x);
    atomicAdd(dst + 5, v * f1.y);
    atomicAdd(dst + 6, v * f1.z);
    atomicAdd(dst + 7, v * f1.w);
}

// ---------------------------------------------------------------------------
// Epilogue: X = PReLU(C + b), float4 per thread.
// ---------------------------------------------------------------------------
__global__ __launch_bounds__(256) void gcn_epilogue(const float* __restrict__ C,
                                                    const float* __restrict__ bias,
                                                    const float* __restrict__ alpha_p,
                                                    float* __restrict__ X,
                                                    int total4) {
    const int i = blockIdx.x * blockDim.x + threadIdx.x;
    if (i >= total4) return;
    const float alpha = alpha_p[0];
    const int bcol = (i & (GCN_D / 4 - 1)) * 4;  // element column within row

    const float4 d = *reinterpret_cast<const float4*>(C + (size_t)i * 4);
    const float4 bb = *reinterpret_cast<const float4*>(bias + bcol);
    float4 o;
    o.x = d.x + bb.x; o.x = o.x >= 0.f ? o.x : alpha * o.x;
    o.y = d.y + bb.y; o.y = o.y >= 0.f ? o.y : alpha * o.y;
    o.z = d.z + bb.z; o.z = o.z >= 0.f ? o.z : alpha * o.z;
    o.w = d.w + bb.w; o.w = o.w >= 0.f ? o.w : alpha * o.w;
    *reinterpret_cast<float4*>(X + (size_t)i * 4) = o;
}

extern "C" void kernel_launch(void* const* d_in, const int* in_sizes, int n_in,
                              void* d_out, int out_size, void* d_ws, size_t ws_size,
                              hipStream_t stream) {
    const float* seq   = (const float*)d_in[0];
    const int*   erows = (const int*)  d_in[1];
    const int*   ecols = (const int*)  d_in[2];
    const float* evals = (const float*)d_in[3];
    const float* W     = (const float*)d_in[4];
    const float* b     = (const float*)d_in[5];
    const float* alpha = (const float*)d_in[6];

    const int n_nodes = in_sizes[0] / GCN_D;
    const int n_edges = in_sizes[1];
    const size_t nd = (size_t)n_nodes * GCN_D;

    float* F = (float*)d_ws;        // fts buffer  [N, D]
    float* C = F + nd;              // accumulator [N, D]
    float* X = (float*)d_out;       // per-layer node features (final = layer 3 output)

    const int gemm_blocks = (n_nodes + ROWS_PER_BLOCK - 1) / ROWS_PER_BLOCK;
    const int spmm_blocks = (n_edges + 7) / 8;
    const int total4      = (int)(nd / 4);
    const int ew_blocks   = (total4 + 255) / 256;

    for (int layer = 0; layer < 3; ++layer) {
        const float* xin = (layer == 0) ? seq : (const float*)X;
        gcn_gemm_wmma<<<gemm_blocks, 256, 0, stream>>>(xin, W + (size_t)layer * GCN_D * GCN_D, F, n_nodes);
        gcn_zero<<<ew_blocks, 256, 0, stream>>>(C, total4);
        gcn_spmm<<<spmm_blocks, 256, 0, stream>>>(erows, ecols, evals, F, C, n_edges);
        gcn_epilogue<<<ew_blocks, 256, 0, stream>>>(C, b + (size_t)layer * GCN_D, alpha + layer, X, total4);
    }
}